// CrossEntropyLoss_9758165696829
// MI455X (gfx1250) — compile-verified
//
#include <hip/hip_runtime.h>

#ifndef __has_builtin
#define __has_builtin(x) 0
#endif

#if __has_builtin(__builtin_amdgcn_global_load_async_to_lds_b128)
#define HAVE_ASYNC_LDS 1
#else
#define HAVE_ASYNC_LDS 0
#endif

namespace {
constexpr int kB = 16;
constexpr int kS = 512;                                   // shifted seq length
constexpr int kSP1 = 513;
constexpr int kV = 32000;
constexpr int kNF4 = kV / 4;                              // 8000 float4 per row
constexpr int kThreads = 256;                             // 8 waves (wave32)
constexpr int kChunkF4 = 1024;                            // 16 KB chunk
constexpr int kNChunk = (kNF4 + kChunkF4 - 1) / kChunkF4; // 8 (last = 832)
constexpr int kLastF4 = kNF4 - (kNChunk - 1) * kChunkF4;  // 832
constexpr int kRows = kB * kS;                            // 8192
}  // namespace

// Online logsumexp update: exactly one v_exp_f32 per element.
// if x>m:  s = s*exp(m-x) + 1 ; m = x      (note m-x = -|x-m|)
// else:    s = s + exp(x-m)                (note x-m = -|x-m|)
__device__ __forceinline__ void online_update(float x, float& m, float& s) {
  float d = x - m;
  float e = __expf(-__builtin_fabsf(d));
  if (d > 0.0f) {
    s = __builtin_fmaf(s, e, 1.0f);
    m = x;
  } else {
    s += e;
  }
}

#if HAVE_ASYNC_LDS
// Builtin signature (from hipcc diagnostic): pointee type is a 16-byte
// gcc-style int vector, AS(1) for the global source, AS(3) for the LDS dest.
typedef int v4i_t __attribute__((vector_size(16)));
typedef __attribute__((address_space(1))) v4i_t* as1_v4i;
typedef __attribute__((address_space(3))) v4i_t* as3_v4i;

__device__ __forceinline__ void wait_async_all() {
  // Wait for all of this wave's async global->LDS transfers, and for any
  // in-flight ds_loads, before the LDS buffer is touched/reused.
  asm volatile("s_wait_asynccnt 0x0\n\ts_wait_dscnt 0x0" ::: "memory");
}

// Each thread async-copies its own float4s: thread t owns indices i*256+t.
// Because it later reads exactly the same LDS slots it wrote, no cross-wave
// barrier is needed; s_wait_asynccnt 0 alone establishes the dependency.
__device__ __forceinline__ void issue_chunk(const float4* __restrict__ gp,
                                            float4* lbuf, int c, int tid) {
  const int base = c * kChunkF4;
  const int nf4 = (c == kNChunk - 1) ? kLastF4 : kChunkF4;
  float4* dst = lbuf + (c & 1) * kChunkF4;
#pragma unroll
  for (int i = 0; i < kChunkF4 / kThreads; ++i) {
    const int idx = i * kThreads + tid;
    if (idx < nf4) {
      __builtin_amdgcn_global_load_async_to_lds_b128(
          (as1_v4i)(gp + base + idx), (as3_v4i)(dst + idx), 0, 0);
    }
  }
}
#endif  // HAVE_ASYNC_LDS

__global__ __launch_bounds__(kThreads) void ce_row_kernel(
    const float* __restrict__ logits_full, const long long* __restrict__ trg,
    const long long* __restrict__ lengths, float* __restrict__ ws_nll,
    int* __restrict__ ws_cnt) {
  const int row = blockIdx.x;         // row = b*512 + s, s in [0,512)
  const int b = row >> 9;
  const int s = row & (kS - 1);
  const int tid = threadIdx.x;

  const long long tgt = trg[(size_t)b * kSP1 + (s + 1)];
  const long long len = lengths[b];
  const bool valid = (s < (int)len) && (tgt != 0);
  if (!valid) {  // masked row: contributes 0, skip the 128 KB read entirely
    if (tid == 0) {
      ws_nll[row] = 0.0f;
      ws_cnt[row] = 0;
    }
    return;
  }

  const float* __restrict__ rowp =
      logits_full + ((size_t)b * kSP1 + (size_t)(s + 1)) * (size_t)kV;
  const float xt = rowp[(int)tgt];

  float m = -__builtin_inff();
  float ssum = 0.0f;

#if HAVE_ASYNC_LDS
  __shared__ float4 buf[2 * kChunkF4];  // 32 KB double buffer
  const float4* gp = (const float4*)rowp;
  issue_chunk(gp, buf, 0, tid);
  for (int c = 0; c < kNChunk; ++c) {
    wait_async_all();                                 // chunk c landed in LDS
    if (c + 1 < kNChunk) issue_chunk(gp, buf, c + 1, tid);  // prefetch next
    const int nf4 = (c == kNChunk - 1) ? kLastF4 : kChunkF4;
    const float4* lb = buf + (c & 1) * kChunkF4;
#pragma unroll
    for (int i = 0; i < kChunkF4 / kThreads; ++i) {
      const int idx = i * kThreads + tid;
      if (idx < nf4) {
        float4 v = lb[idx];  // ds_load_b128 of this thread's own staged data
        online_update(v.x, m, ssum);
        online_update(v.y, m, ssum);
        online_update(v.z, m, ssum);
        online_update(v.w, m, ssum);
      }
    }
  }
#else
  const float4* gp = (const float4*)rowp;
  for (int f = tid; f < kNF4; f += kThreads) {
    float4 v = gp[f];
    online_update(v.x, m, ssum);
    online_update(v.y, m, ssum);
    online_update(v.z, m, ssum);
    online_update(v.w, m, ssum);
  }
#endif

  // Block-wide merge of (m, s) pairs: M = max, S = s1*e^(m1-M) + s2*e^(m2-M).
  __shared__ float rm[kThreads];
  __shared__ float rs[kThreads];
  rm[tid] = m;
  rs[tid] = ssum;
  __syncthreads();
#pragma unroll
  for (int off = kThreads / 2; off > 0; off >>= 1) {
    if (tid < off) {
      const float m1 = rm[tid], m2 = rm[tid + off];
      const float s1 = rs[tid], s2 = rs[tid + off];
      const float M = fmaxf(m1, m2);
      rs[tid] = s1 * __expf(m1 - M) + s2 * __expf(m2 - M);
      rm[tid] = M;
    }
    __syncthreads();
  }
  if (tid == 0) {
    const float lse = rm[0] + __logf(rs[0]);
    ws_nll[row] = lse - xt;  // nll = logsumexp - x_target
    ws_cnt[row] = 1;
  }
}

// Deterministic final reduction: fixed strided order + fixed tree, no atomics.
__global__ __launch_bounds__(kThreads) void ce_reduce_kernel(
    const float* __restrict__ ws_nll, const int* __restrict__ ws_cnt,
    float* __restrict__ out) {
  __shared__ float sf[kThreads];
  __shared__ int si[kThreads];
  float acc = 0.0f;
  int cnt = 0;
  for (int i = threadIdx.x; i < kRows; i += kThreads) {
    acc += ws_nll[i];
    cnt += ws_cnt[i];
  }
  sf[threadIdx.x] = acc;
  si[threadIdx.x] = cnt;
  __syncthreads();
#pragma unroll
  for (int off = kThreads / 2; off > 0; off >>= 1) {
    if (threadIdx.x < off) {
      sf[threadIdx.x] += sf[threadIdx.x + off];
      si[threadIdx.x] += si[threadIdx.x + off];
    }
    __syncthreads();
  }
  if (threadIdx.x == 0) {
    const int nv = si[0] < 1 ? 1 : si[0];
    out[0] = sf[0] / (float)nv;
  }
}

extern "C" void kernel_launch(void* const* d_in, const int* in_sizes, int n_in,
                              void* d_out, int out_size, void* d_ws,
                              size_t ws_size, hipStream_t stream) {
  (void)in_sizes;
  (void)n_in;
  (void)out_size;
  (void)ws_size;
  const float* logits = (const float*)d_in[0];         // (16, 513, 32000) f32
  const long long* trg = (const long long*)d_in[1];    // (16, 513) i64
  const long long* lens = (const long long*)d_in[2];   // (16,) i64

  float* ws_nll = (float*)d_ws;                        // 8192 floats
  int* ws_cnt = (int*)((char*)d_ws + kRows * sizeof(float));  // 8192 ints

  ce_row_kernel<<<dim3(kRows), dim3(kThreads), 0, stream>>>(logits, trg, lens,
                                                            ws_nll, ws_cnt);
  ce_reduce_kernel<<<dim3(1), dim3(kThreads), 0, stream>>>(
      ws_nll, ws_cnt, (float*)d_out);
}